// TransModel_81054622810357
// MI455X (gfx1250) — compile-verified
//
#include <hip/hip_runtime.h>

#define BS 8
#define NUM_ACT 8
#define NUM_OBJ 15
#define D 256
#define L 16
#define S 23
#define NHEAD 8
#define HEAD_DIM 32
#define DFF 2048

typedef float v2f __attribute__((ext_vector_type(2)));
typedef float v8f __attribute__((ext_vector_type(8)));

__device__ inline v8f wmma4(v2f a, v2f b, v8f c) {
  return __builtin_amdgcn_wmma_f32_16x16x4_f32(false, a, false, b, (short)0, c, false, false);
}

// ---------------------------------------------------------------------------
// Generic GEMM: C[M,N] = act( A[M,K] @ W[N,K]^T + bias[N] )
// One wave per 32x64 macro-tile (2 M-tiles x 4 N-tiles of 16x16), so each
// k-step issues 6 b64 loads and 8 WMMAs. 8 waves per 256-thread block.
// M multiple of 32, N multiple of 64, K multiple of 4.
// ---------------------------------------------------------------------------
__global__ void gemm_wmma_kernel(const float* __restrict__ A,
                                 const float* __restrict__ W,
                                 const float* __restrict__ bias,
                                 float* __restrict__ C,
                                 int M, int N, int K, int relu) {
  int tilesN = N >> 6;                       // 64-wide macro columns
  int nTiles = (M >> 5) * tilesN;            // 32-tall macro rows
  int tile = blockIdx.x * 8 + (threadIdx.x >> 5);
  if (tile >= nTiles) return;                // uniform per-wave exit
  int m0 = (tile / tilesN) << 5;
  int n0 = (tile % tilesN) << 6;
  int lane = threadIdx.x & 31;
  int half = lane >> 4;
  int l = lane & 15;

  const float* __restrict__ A0 = A + (size_t)(m0 + l) * K;        // rows m0..m0+15
  const float* __restrict__ A1 = A0 + (size_t)16 * K;             // rows m0+16..m0+31
  const float* __restrict__ W0 = W + (size_t)(n0 + l) * K;        // B[k][n] = W[n][k]

  v8f acc[2][4] = {};
  for (int k = 0; k < K; k += 4) {
    int kk = k + 2 * half;
    v2f a0, a1, b[4];
    a0.x = A0[kk]; a0.y = A0[kk + 1];
    a1.x = A1[kk]; a1.y = A1[kk + 1];
#pragma unroll
    for (int j = 0; j < 4; ++j) {
      b[j].x = W0[(size_t)j * 16 * K + kk];
      b[j].y = W0[(size_t)j * 16 * K + kk + 1];
    }
#pragma unroll
    for (int j = 0; j < 4; ++j) {
      acc[0][j] = wmma4(a0, b[j], acc[0][j]);
      acc[1][j] = wmma4(a1, b[j], acc[1][j]);
    }
  }
#pragma unroll
  for (int i = 0; i < 2; ++i)
#pragma unroll
    for (int j = 0; j < 4; ++j) {
      float bn = bias ? bias[n0 + j * 16 + l] : 0.0f;
#pragma unroll
      for (int r = 0; r < 8; ++r) {
        int row = m0 + i * 16 + r + 8 * half;
        float v = acc[i][j][r] + bn;
        if (relu) v = fmaxf(v, 0.0f);
        C[(size_t)row * N + n0 + j * 16 + l] = v;
      }
    }
}

// ---------------------------------------------------------------------------
// Attention: one wave per (batch, head). T=16, HEAD_DIM=32.
// scores(16x16,K=32) -> softmax -> context(16x32,K=16), all via WMMA.
// qkv layout: [b*16+t][768] with q|k|v at offsets 0|256|512, head h at h*32.
// ---------------------------------------------------------------------------
__global__ void attn_kernel(const float* __restrict__ qkv, float* __restrict__ o) {
  int b = blockIdx.x / NHEAD;
  int h = blockIdx.x % NHEAD;
  int lane = threadIdx.x & 31;
  int half = lane >> 4;
  int l = lane & 15;
  __shared__ float att[16][16];
  const float* __restrict__ base = qkv + (size_t)b * 16 * 768;
  const int hb = h * 32;

  v8f acc = {};
  for (int k = 0; k < 32; k += 4) {
    v2f a, bb;
    a.x = base[l * 768 + hb + k + 2 * half];            // q[m=l][k]
    a.y = base[l * 768 + hb + k + 2 * half + 1];
    bb.x = base[l * 768 + 256 + hb + k + 2 * half];     // k[n=l][k]
    bb.y = base[l * 768 + 256 + hb + k + 2 * half + 1];
    acc = wmma4(a, bb, acc);
  }
  const float scale = 0.1767766952966369f;              // 1/sqrt(32)
#pragma unroll
  for (int r = 0; r < 8; ++r) {
    float s = acc[r] * scale;
    float mx = s;
    for (int msk = 1; msk < 16; msk <<= 1) mx = fmaxf(mx, __shfl_xor(mx, msk, 32));
    float e = __expf(s - mx);
    float sm = e;
    for (int msk = 1; msk < 16; msk <<= 1) sm += __shfl_xor(sm, msk, 32);
    att[r + 8 * half][l] = e / sm;
  }
  __syncthreads();
  for (int nt = 0; nt < 2; ++nt) {
    v8f oacc = {};
    for (int k = 0; k < 16; k += 4) {
      v2f a, bb;
      a.x = att[l][k + 2 * half];
      a.y = att[l][k + 2 * half + 1];
      bb.x = base[(k + 2 * half) * 768 + 512 + hb + nt * 16 + l];   // v[k][n]
      bb.y = base[(k + 2 * half + 1) * 768 + 512 + hb + nt * 16 + l];
      oacc = wmma4(a, bb, oacc);
    }
#pragma unroll
    for (int r = 0; r < 8; ++r)
      o[(size_t)(b * 16 + r + 8 * half) * D + hb + nt * 16 + l] = oacc[r];
  }
}

// x = LayerNorm(x + r) * g + b ; one block per token, 256 threads.
__global__ void add_ln_kernel(float* __restrict__ x, const float* __restrict__ r,
                              const float* __restrict__ g, const float* __restrict__ bta) {
  int tkn = blockIdx.x;
  int d = threadIdx.x;
  __shared__ float red[256];
  float v = x[tkn * D + d] + r[tkn * D + d];
  red[d] = v; __syncthreads();
  for (int s = 128; s > 0; s >>= 1) { if (d < s) red[d] += red[d + s]; __syncthreads(); }
  float mean = red[0] * (1.0f / D);
  __syncthreads();
  float c = v - mean;
  red[d] = c * c; __syncthreads();
  for (int s = 128; s > 0; s >>= 1) { if (d < s) red[d] += red[d + s]; __syncthreads(); }
  float var = red[0] * (1.0f / D);
  x[tkn * D + d] = c * rsqrtf(var + 1e-5f) * g[d] + bta[d];
}

// ot = src[:, :16]; ans[:,0] = ot
__global__ void init_kernel(float* __restrict__ ot, float* __restrict__ ans,
                            const float* __restrict__ src) {
  int i = blockIdx.x * 256 + threadIdx.x;       // 8*16*256
  int b = i / (L * D), rest = i % (L * D);
  float v = src[(size_t)b * S * D + rest];
  ot[i] = v;
  ans[(size_t)(b * 9) * L * D + rest] = v;
}

// ot[:, 15, :] = src[:, 15+t, :]
__global__ void set_act_kernel(float* __restrict__ ot, const float* __restrict__ src, int t) {
  int b = blockIdx.x, d = threadIdx.x;
  ot[(b * L + (L - 1)) * D + d] = src[((size_t)b * S + (L - 1) + t) * D + d];
}

// ans[:, t+1] = ot
__global__ void store_ans_kernel(const float* __restrict__ ot, float* __restrict__ ans, int t) {
  int i = blockIdx.x * 256 + threadIdx.x;
  int b = i / (L * D), rest = i % (L * D);
  ans[((size_t)(b * 9) + t + 1) * L * D + rest] = ot[i];
}

// ---------------------------------------------------------------------------
// D1: per batch — aff FC1 (relu), seg FC1 (relu), view head. 512 threads.
// ---------------------------------------------------------------------------
__global__ void dec_heads_kernel(const float* __restrict__ ot, const float* __restrict__ src,
                                 const int* __restrict__ seg_id,
                                 const float* __restrict__ afw1, const float* __restrict__ afb1,
                                 const float* __restrict__ sgw1, const float* __restrict__ sgb1,
                                 const float* __restrict__ vw1, const float* __restrict__ vb1,
                                 const float* __restrict__ vw2, const float* __restrict__ vb2,
                                 float* __restrict__ affh, float* __restrict__ segh,
                                 float* __restrict__ view_out, int t) {
  int b = blockIdx.x;
  int tid = threadIdx.x;   // 512
  __shared__ float hcat[512];
  __shared__ float vh[256];
  int tg = min(max(seg_id[b * NUM_ACT + t], 0), NUM_OBJ - 1);
  if (tid < 256) hcat[tid] = ot[(b * L + tg) * D + tid];                           // tok
  else           hcat[tid] = src[((size_t)b * S + (L - 1) + t) * D + (tid - 256)]; // xa
  __syncthreads();
  { // affh = relu(aff_fc1 @ [tok;xa] + b)
    float s = afb1[tid];
    for (int k = 0; k < 512; ++k) s += afw1[tid * 512 + k] * hcat[k];
    affh[b * 512 + tid] = fmaxf(s, 0.0f);
  }
  { // segh = relu(seg_fc1 @ tok + b)
    float s = sgb1[tid];
    for (int k = 0; k < 256; ++k) s += sgw1[tid * 256 + k] * hcat[k];
    segh[b * 512 + tid] = fmaxf(s, 0.0f);
  }
  if (tid < 256) { // view hidden
    float s = vb1[tid];
    for (int k = 0; k < 256; ++k) s += vw1[tid * 256 + k] * hcat[k];
    vh[tid] = fmaxf(s, 0.0f);
  }
  __syncthreads();
  if (tid < 2) {
    float s = vb2[tid];
    for (int k = 0; k < 256; ++k) s += vw2[tid * 256 + k] * vh[k];
    view_out[(b * NUM_ACT + t) * 2 + tid] = s;   // [1,b,t,1,2]
  }
}

// ---------------------------------------------------------------------------
// D2: seg decoder, fused FC2 -> tconv1(relu) -> tconv2 -> bilinear 64->128.
// One block per batch, 256 threads, z kept in LDS (2x64x64 = 32KB).
// ---------------------------------------------------------------------------
__global__ void seg_dec_kernel(const float* __restrict__ segh,
                               const float* __restrict__ w2, const float* __restrict__ b2,
                               const float* __restrict__ tc1w, const float* __restrict__ tc1b,
                               const float* __restrict__ tc2w, const float* __restrict__ tc2b,
                               float* __restrict__ seg_out, int t) {
  int b = blockIdx.x;
  int tid = threadIdx.x;
  __shared__ float sh[512];
  __shared__ float s16[1024];        // (4,16,16)
  __shared__ float z[2 * 64 * 64];   // (2,64,64)
  sh[tid] = segh[b * 512 + tid];
  sh[tid + 256] = segh[b * 512 + 256 + tid];
  __syncthreads();
  for (int i = tid; i < 1024; i += 256) {
    float s = b2[i];
    for (int k = 0; k < 512; ++k) s += w2[i * 512 + k] * sh[k];
    s16[i] = s;
  }
  __syncthreads();
  for (int pos = tid; pos < 1024; pos += 256) {
    int u = pos >> 5, v = pos & 31;          // 32x32 position
    int ci = (u >> 1) * 16 + (v >> 1);       // 16x16 pixel
    int p = u & 1, q = v & 1;
    float m[64];
    for (int dch = 0; dch < 64; ++dch) {
      float s = tc1b[dch];
      for (int c = 0; c < 4; ++c)
        s += s16[c * 256 + ci] * tc1w[((c * 64 + dch) * 2 + p) * 2 + q];
      m[dch] = fmaxf(s, 0.0f);
    }
    for (int e = 0; e < 2; ++e)
      for (int dy = 0; dy < 2; ++dy)
        for (int dx = 0; dx < 2; ++dx) {
          float s = tc2b[e];
          for (int dch = 0; dch < 64; ++dch)
            s += m[dch] * tc2w[((dch * 2 + e) * 2 + dy) * 2 + dx];
          z[(e * 64 + (2 * u + dy)) * 64 + (2 * v + dx)] = s;
        }
  }
  __syncthreads();
  const float f = 63.0f / 127.0f;   // align_corners-style up2
  for (int i = tid; i < 128 * 128; i += 256) {
    int r = i >> 7, c = i & 127;
    float cr = r * f; int i0 = (int)floorf(cr); int i1 = min(i0 + 1, 63); float tr = cr - i0;
    float cc = c * f; int j0 = (int)floorf(cc); int j1 = min(j0 + 1, 63); float tcc = cc - j0;
    for (int e = 0; e < 2; ++e) {
      float v00 = z[(e * 64 + i0) * 64 + j0], v01 = z[(e * 64 + i0) * 64 + j1];
      float v10 = z[(e * 64 + i1) * 64 + j0], v11 = z[(e * 64 + i1) * 64 + j1];
      float top = v00 * (1 - tcc) + v01 * tcc;
      float bot = v10 * (1 - tcc) + v11 * tcc;
      seg_out[((((size_t)b * NUM_ACT + t) * 128 + r) * 128 + c) * 2 + e] = top * (1 - tr) + bot * tr;
    }
  }
}

// ---------------------------------------------------------------------------
// D3: affordance gathered pixel only -> new_tok; writes ot[b, tgt].
// One block per batch, 256 threads (one per output channel).
// ---------------------------------------------------------------------------
__global__ void aff_tok_kernel(const float* __restrict__ affh,
                               const float* __restrict__ w2, const float* __restrict__ b2,
                               const float* __restrict__ tc1w, const float* __restrict__ tc1b,
                               const float* __restrict__ tc2w, const float* __restrict__ tc2b,
                               const int* __restrict__ sel, const int* __restrict__ seg_id,
                               float* __restrict__ ot, int t) {
  int b = blockIdx.x;
  int tid = threadIdx.x;   // 256
  __shared__ float h[512];
  __shared__ float v4[4];
  __shared__ float m[128];
  h[tid] = affh[b * 512 + tid];
  h[tid + 256] = affh[b * 512 + 256 + tid];
  __syncthreads();
  int r = min(max(sel[(b * NUM_ACT + t) * 2 + 0], 0), 127);
  int c = min(max(sel[(b * NUM_ACT + t) * 2 + 1], 0), 127);
  const float f = 63.0f / 127.0f;
  float cr = r * f; int i0 = (int)floorf(cr); int i1 = min(i0 + 1, 63); float tr = cr - i0;
  float cc = c * f; int j0 = (int)floorf(cc); int j1 = min(j0 + 1, 63); float tcc = cc - j0;
  int ys[2] = {i0, i1}, xs[2] = {j0, j1};
  float wy[2] = {1.0f - tr, tr}, wx[2] = {1.0f - tcc, tcc};
  float acc = 0.0f;
  for (int a = 0; a < 2; ++a)
    for (int bb = 0; bb < 2; ++bb) {
      int y = ys[a], x = xs[bb];
      float wgt = wy[a] * wx[bb];
      int u = y >> 1, v = x >> 1;             // 32x32
      int ci = (u >> 1) * 16 + (v >> 1);      // 16x16
      int p = u & 1, q = v & 1;
      int dy = y & 1, dx = x & 1;
      __syncthreads();
      if (tid < 4) {                          // 4 rows of FC2
        float s = b2[tid * 256 + ci];
        for (int k = 0; k < 512; ++k) s += w2[(tid * 256 + ci) * 512 + k] * h[k];
        v4[tid] = s;
      }
      __syncthreads();
      if (tid < 128) {                        // tconv1 pixel, 128 ch, relu
        float s = tc1b[tid];
        for (int ch = 0; ch < 4; ++ch) s += v4[ch] * tc1w[((ch * 128 + tid) * 2 + p) * 2 + q];
        m[tid] = fmaxf(s, 0.0f);
      }
      __syncthreads();
      float s = tc2b[tid];                    // tconv2 pixel, 256 ch
      for (int dch = 0; dch < 128; ++dch)
        s += m[dch] * tc2w[((dch * 256 + tid) * 2 + dy) * 2 + dx];
      acc += wgt * s;
    }
  int tg = min(max(seg_id[b * NUM_ACT + t], 0), NUM_OBJ - 1);
  ot[(b * L + tg) * D + tid] = acc;
}

// ---------------------------------------------------------------------------
extern "C" void kernel_launch(void* const* d_in, const int* in_sizes, int n_in,
                              void* d_out, int out_size, void* d_ws, size_t ws_size,
                              hipStream_t stream) {
  const float* src      = (const float*)d_in[0];
  const int*   aff_sel  = (const int*)d_in[2];
  const int*   seg_id   = (const int*)d_in[3];
  const float* aff_fc1w = (const float*)d_in[6];
  const float* aff_fc1b = (const float*)d_in[7];
  const float* aff_fc2w = (const float*)d_in[8];
  const float* aff_fc2b = (const float*)d_in[9];
  const float* aff_tc1w = (const float*)d_in[10];
  const float* aff_tc1b = (const float*)d_in[11];
  const float* aff_tc2w = (const float*)d_in[12];
  const float* aff_tc2b = (const float*)d_in[13];
  const float* seg_fc1w = (const float*)d_in[14];
  const float* seg_fc1b = (const float*)d_in[15];
  const float* seg_fc2w = (const float*)d_in[16];
  const float* seg_fc2b = (const float*)d_in[17];
  const float* seg_tc1w = (const float*)d_in[18];
  const float* seg_tc1b = (const float*)d_in[19];
  const float* seg_tc2w = (const float*)d_in[20];
  const float* seg_tc2b = (const float*)d_in[21];
  const float* view_w1  = (const float*)d_in[22];
  const float* view_b1  = (const float*)d_in[23];
  const float* view_w2  = (const float*)d_in[24];
  const float* view_b2  = (const float*)d_in[25];
  const float* enc_in_w  = (const float*)d_in[26];
  const float* enc_in_b  = (const float*)d_in[27];
  const float* enc_out_w = (const float*)d_in[28];
  const float* enc_out_b = (const float*)d_in[29];
  const float* enc_ff1_w = (const float*)d_in[30];
  const float* enc_ff1_b = (const float*)d_in[31];
  const float* enc_ff2_w = (const float*)d_in[32];
  const float* enc_ff2_b = (const float*)d_in[33];
  const float* ln1_g = (const float*)d_in[34];
  const float* ln1_b = (const float*)d_in[35];
  const float* ln2_g = (const float*)d_in[36];
  const float* ln2_b = (const float*)d_in[37];

  float* ans      = (float*)d_out;                 // (8,9,16,256)
  float* seg_out  = ans + (size_t)BS * 9 * L * D;  // (1,8,8,128,128,2)
  float* view_out = seg_out + (size_t)BS * NUM_ACT * 128 * 128 * 2; // (1,8,8,1,2)

  float* ws   = (float*)d_ws;
  float* ot   = ws;                 ws += BS * L * D;        // 32768
  float* qkv  = ws;                 ws += BS * L * 3 * D;    // 98304
  float* attO = ws;                 ws += BS * L * D;        // 32768
  float* proj = ws;                 ws += BS * L * DFF;      // 262144
  float* affh = ws;                 ws += BS * 512;
  float* segh = ws;                 ws += BS * 512;
  float* f1   = proj;               // FF hidden reuses the big buffer
  float* f2   = qkv;                // FF output reuses qkv buffer (qkv dead by then)

  const int M = BS * L;  // 128 tokens

  init_kernel<<<(BS * L * D) / 256, 256, 0, stream>>>(ot, ans, src);

  for (int t = 0; t < NUM_ACT; ++t) {
    set_act_kernel<<<BS, D, 0, stream>>>(ot, src, t);
    dec_heads_kernel<<<BS, 512, 0, stream>>>(ot, src, seg_id,
        aff_fc1w, aff_fc1b, seg_fc1w, seg_fc1b,
        view_w1, view_b1, view_w2, view_b2, affh, segh, view_out, t);
    seg_dec_kernel<<<BS, 256, 0, stream>>>(segh, seg_fc2w, seg_fc2b,
        seg_tc1w, seg_tc1b, seg_tc2w, seg_tc2b, seg_out, t);
    aff_tok_kernel<<<BS, 256, 0, stream>>>(affh, aff_fc2w, aff_fc2b,
        aff_tc1w, aff_tc1b, aff_tc2w, aff_tc2b, aff_sel, seg_id, ot, t);

    for (int lyr = 0; lyr < 2; ++lyr) {
      const float* inw  = enc_in_w  + (size_t)lyr * 3 * D * D;
      const float* inb  = enc_in_b  + (size_t)lyr * 3 * D;
      const float* outw = enc_out_w + (size_t)lyr * D * D;
      const float* outb = enc_out_b + (size_t)lyr * D;
      const float* f1w  = enc_ff1_w + (size_t)lyr * DFF * D;
      const float* f1b  = enc_ff1_b + (size_t)lyr * DFF;
      const float* f2w  = enc_ff2_w + (size_t)lyr * D * DFF;
      const float* f2b  = enc_ff2_b + (size_t)lyr * D;

      // qkv = ot @ in_w^T + in_b            (128 x 768, K=256): 48 macro-tiles
      gemm_wmma_kernel<<<(48 + 7) / 8, 256, 0, stream>>>(ot, inw, inb, qkv, M, 3 * D, D, 0);
      // attention per (b, head)
      attn_kernel<<<BS * NHEAD, 32, 0, stream>>>(qkv, attO);
      // proj = attO @ out_w^T + out_b       (128 x 256, K=256): 16 macro-tiles
      gemm_wmma_kernel<<<(16 + 7) / 8, 256, 0, stream>>>(attO, outw, outb, proj, M, D, D, 0);
      add_ln_kernel<<<M, D, 0, stream>>>(ot, proj, ln1_g + lyr * D, ln1_b + lyr * D);
      // f1 = relu(ot @ ff1^T + b)           (128 x 2048, K=256): 128 macro-tiles
      gemm_wmma_kernel<<<(128 + 7) / 8, 256, 0, stream>>>(ot, f1w, f1b, f1, M, DFF, D, 1);
      // f2 = f1 @ ff2^T + b                 (128 x 256, K=2048): 16 macro-tiles
      gemm_wmma_kernel<<<(16 + 7) / 8, 256, 0, stream>>>(f1, f2w, f2b, f2, M, D, DFF, 0);
      add_ln_kernel<<<M, D, 0, stream>>>(ot, f2, ln2_g + lyr * D, ln2_b + lyr * D);
    }
    store_ans_kernel<<<(BS * L * D) / 256, 256, 0, stream>>>(ot, ans, t);
  }
}